// LightGCN_32942399160713
// MI455X (gfx1250) — compile-verified
//
#include <hip/hip_runtime.h>

#define N_USERS  100000
#define N_ITEMS  50000
#define N_NODES  150000
#define EMB_DIM  64
#define NNZ_E    4800000
#define N_LAYERS 3
#define CHUNK    256   // edge pairs staged per async chunk (2 KB)

typedef __attribute__((ext_vector_type(2))) float v2f;
typedef __attribute__((ext_vector_type(8))) float v8f;

// Low 32 bits of a generic (flat) LDS address are the LDS byte offset
// (ISA 10.2: LDS aperture).
__device__ __forceinline__ unsigned lds_lo(const void* p) {
  return (unsigned)(unsigned long long)p;
}

// ---------------- CSR build ----------------

__global__ void zero_u32_kernel(unsigned* a, int n) {
  int i = blockIdx.x * blockDim.x + threadIdx.x;
  if (i < n) a[i] = 0u;
}

__global__ void hist_kernel(const int* __restrict__ rows, unsigned* __restrict__ cnt) {
  int i = blockIdx.x * blockDim.x + threadIdx.x;
  if (i < NNZ_E) atomicAdd(&cnt[rows[i]], 1u);
}

__global__ __launch_bounds__(1024) void scan_kernel(unsigned* __restrict__ cnt,
                                                    unsigned* __restrict__ row_ptr) {
  __shared__ unsigned part[1024];
  const int t = threadIdx.x;
  const int chunk = (N_NODES + 1023) / 1024;
  const int lo = t * chunk;
  const int hi = min(lo + chunk, N_NODES);
  unsigned s = 0;
  for (int i = lo; i < hi; ++i) s += cnt[i];
  part[t] = s;
  __syncthreads();
  for (int o = 1; o < 1024; o <<= 1) {
    unsigned v = (t >= o) ? part[t - o] : 0u;
    __syncthreads();
    part[t] += v;
    __syncthreads();
  }
  unsigned run = (t == 0) ? 0u : part[t - 1];
  for (int i = lo; i < hi; ++i) {
    unsigned c = cnt[i];
    row_ptr[i] = run;   // CSR row pointer
    cnt[i] = run;       // scatter cursor (destroyed by scatter_kernel)
    run += c;
  }
  if (t == 1023) row_ptr[N_NODES] = run;
}

// pair word: [63:32] = f32 weight bits, [31:28] = dest row & 15, [27:0] = col (<150000)
__global__ void scatter_kernel(const int* __restrict__ rows, const int* __restrict__ cols,
                               const float* __restrict__ vals,
                               const unsigned char* __restrict__ mask,
                               unsigned* __restrict__ cursor,
                               unsigned long long* __restrict__ pairs) {
  int i = blockIdx.x * blockDim.x + threadIdx.x;
  if (i >= NNZ_E) return;
  unsigned r = (unsigned)rows[i];
  unsigned pos = atomicAdd(&cursor[r], 1u);
  // sparse dropout: kept edges scaled by 1/KEEP_PROB = 2, dropped -> 0
  float v = mask[i] ? vals[i] * 2.0f : 0.0f;
  unsigned lo = (unsigned)cols[i] | ((r & 15u) << 28);
  pairs[pos] = ((unsigned long long)__float_as_uint(v) << 32) | lo;
}

// ---------------- init ----------------

__global__ void init_kernel(const float* __restrict__ uw, const float* __restrict__ iw,
                            float* __restrict__ embA, float* __restrict__ acc) {
  int i = blockIdx.x * blockDim.x + threadIdx.x;
  if (i >= N_NODES * EMB_DIM) return;
  float x = (i < N_USERS * EMB_DIM) ? uw[i] : iw[i - N_USERS * EMB_DIM];
  embA[i] = x;
  acc[i]  = x;
}

// ---------------- WMMA SpMM: one wave owns 16 output rows ----------------

template <int WRITE_NEXT>
__global__ __launch_bounds__(32) void spmm_wmma_kernel(
    const float* __restrict__ emb, float* __restrict__ embN,
    float* __restrict__ acc, const unsigned* __restrict__ row_ptr,
    const unsigned long long* __restrict__ pairs, float fscale) {
  __shared__ unsigned long long sbuf[2][CHUNK];

  const int lane = threadIdx.x;
  const int R = blockIdx.x * 16;                // 16-aligned -> m = row & 15
  const unsigned start = row_ptr[R];            // uniform (scalar) loads
  const unsigned total = row_ptr[R + 16] - start;
  const int nchunks = (int)((total + CHUNK - 1) / CHUNK);

  v8f C0 = {}, C1 = {}, C2 = {}, C3 = {};
  const int half = lane >> 4;     // 0: edges 0/1, 1: edges 2/3 (A/B K-split)
  const int l15  = lane & 15;

  auto issue = [&](int c, int p) {
    const unsigned long long gbase =
        (unsigned long long)(pairs + start + (unsigned long long)c * CHUNK);
#pragma unroll
    for (int q = 0; q < CHUNK / 64; ++q) {
      unsigned voff  = (unsigned)(q * 512 + lane * 16);  // bytes into chunk
      unsigned laddr = lds_lo(&sbuf[p][q * 64 + lane * 2]);
      asm volatile("global_load_async_to_lds_b128 %0, %1, %2 offset:0"
                   :: "v"(laddr), "v"(voff), "s"(gbase) : "memory");
    }
  };

  if (nchunks > 0) issue(0, 0);
  for (int c = 0; c < nchunks; ++c) {
    const int p = c & 1;
    if (c + 1 < nchunks) {
      issue(c + 1, p ^ 1);
      asm volatile("s_wait_asynccnt 4" ::: "memory");  // chunk c complete (in-order)
    } else {
      asm volatile("s_wait_asynccnt 0" ::: "memory");
    }
    const unsigned cbase = (unsigned)c * CHUNK;
    const unsigned nE = min((unsigned)CHUNK, total - cbase);
    const int ngroups = (int)((nE + 3u) / 4u);
    for (int g = 0; g < ngroups; ++g) {
      unsigned col[4]; float val[4]; unsigned m[4];
#pragma unroll
      for (int k = 0; k < 4; ++k) {
        const unsigned rel = cbase + (unsigned)(g * 4 + k);
        const bool ok = rel < total;
        const unsigned long long pk = sbuf[p][g * 4 + k];  // LDS broadcast read
        const unsigned plo = (unsigned)pk;
        col[k] = ok ? (plo & 0x0FFFFFFFu) : 0u;
        m[k]   = plo >> 28;
        val[k] = ok ? __uint_as_float((unsigned)(pk >> 32)) : 0.0f;
      }
      // A (16x4 f32): lanes 0-15 hold K0/K1, lanes 16-31 hold K2/K3; M = l15
      const int ka = half * 2, kb = ka + 1;
      v2f A;
      A.x = (m[ka] == (unsigned)l15) ? val[ka] : 0.0f;
      A.y = (m[kb] == (unsigned)l15) ? val[kb] : 0.0f;
      // B (4x16 f32) per 16-dim tile: row K = gathered embedding row, N = l15
      const float* ra = emb + (unsigned long long)col[ka] * EMB_DIM;
      const float* rb = emb + (unsigned long long)col[kb] * EMB_DIM;
      v2f B0, B1, B2, B3;
      B0.x = ra[ 0 + l15]; B0.y = rb[ 0 + l15];
      B1.x = ra[16 + l15]; B1.y = rb[16 + l15];
      B2.x = ra[32 + l15]; B2.y = rb[32 + l15];
      B3.x = ra[48 + l15]; B3.y = rb[48 + l15];
      C0 = __builtin_amdgcn_wmma_f32_16x16x4_f32(false, A, false, B0, (short)0, C0, false, false);
      C1 = __builtin_amdgcn_wmma_f32_16x16x4_f32(false, A, false, B1, (short)0, C1, false, false);
      C2 = __builtin_amdgcn_wmma_f32_16x16x4_f32(false, A, false, B2, (short)0, C2, false, false);
      C3 = __builtin_amdgcn_wmma_f32_16x16x4_f32(false, A, false, B3, (short)0, C3, false, false);
    }
  }

  // C/D layout: VGPR j -> M = j (lanes 0-15) or j+8 (lanes 16-31), N = l15
  // Straight-line epilogue: gather all acc values first (clause-able), then store.
  float a0[8], a1[8], a2[8], a3[8];
#pragma unroll
  for (int j = 0; j < 8; ++j) {
    const int row = R + j + half * 8;
    const unsigned long long base = (unsigned long long)row * EMB_DIM + (unsigned)l15;
    a0[j] = acc[base +  0];
    a1[j] = acc[base + 16];
    a2[j] = acc[base + 32];
    a3[j] = acc[base + 48];
  }
#pragma unroll
  for (int j = 0; j < 8; ++j) {
    const int row = R + j + half * 8;
    const unsigned long long base = (unsigned long long)row * EMB_DIM + (unsigned)l15;
    acc[base +  0] = (a0[j] + C0[j]) * fscale;
    acc[base + 16] = (a1[j] + C1[j]) * fscale;
    acc[base + 32] = (a2[j] + C2[j]) * fscale;
    acc[base + 48] = (a3[j] + C3[j]) * fscale;
    if (WRITE_NEXT) {
      embN[base +  0] = C0[j];
      embN[base + 16] = C1[j];
      embN[base + 32] = C2[j];
      embN[base + 48] = C3[j];
    }
  }
}

// ---------------- host orchestration ----------------

extern "C" void kernel_launch(void* const* d_in, const int* in_sizes, int n_in,
                              void* d_out, int out_size, void* d_ws, size_t ws_size,
                              hipStream_t stream) {
  (void)in_sizes; (void)n_in; (void)out_size; (void)ws_size;
  const float* user_w = (const float*)d_in[0];
  const float* item_w = (const float*)d_in[1];
  const float* vals   = (const float*)d_in[2];
  const int*   rows   = (const int*)d_in[3];
  const int*   cols   = (const int*)d_in[4];
  const unsigned char* mask = (const unsigned char*)d_in[5];  // jax bool: 1 byte/elt
  float* acc = (float*)d_out;

  char* ws = (char*)d_ws;
  size_t off = 0;
  auto take = [&](size_t bytes) -> char* {
    char* p = ws + off;
    off = (off + bytes + 255) & ~(size_t)255;
    return p;
  };
  unsigned* row_ptr = (unsigned*)take((size_t)(N_NODES + 1) * sizeof(unsigned));
  unsigned* cursor  = (unsigned*)take((size_t)N_NODES * sizeof(unsigned));
  unsigned long long* pairs =
      (unsigned long long*)take((size_t)NNZ_E * 8 + 4096);  // slack: async over-read
  float* embA = (float*)take((size_t)N_NODES * EMB_DIM * sizeof(float));
  float* embB = (float*)take((size_t)N_NODES * EMB_DIM * sizeof(float));

  zero_u32_kernel<<<(N_NODES + 255) / 256, 256, 0, stream>>>(cursor, N_NODES);
  hist_kernel<<<(NNZ_E + 255) / 256, 256, 0, stream>>>(rows, cursor);
  scan_kernel<<<1, 1024, 0, stream>>>(cursor, row_ptr);
  scatter_kernel<<<(NNZ_E + 255) / 256, 256, 0, stream>>>(rows, cols, vals, mask, cursor, pairs);
  init_kernel<<<(N_NODES * EMB_DIM + 255) / 256, 256, 0, stream>>>(user_w, item_w, embA, acc);

  // layers 0..1: acc += layer output, write next-layer embeddings
  spmm_wmma_kernel<1><<<N_NODES / 16, 32, 0, stream>>>(embA, embB, acc, row_ptr, pairs, 1.0f);
  spmm_wmma_kernel<1><<<N_NODES / 16, 32, 0, stream>>>(embB, embA, acc, row_ptr, pairs, 1.0f);
  // last layer: fold final 1/(N_LAYERS+1) scale, skip dead next-layer store
  spmm_wmma_kernel<0><<<N_NODES / 16, 32, 0, stream>>>(embA, embB, acc, row_ptr, pairs, 0.25f);
}